// CentLoss_44143673868455
// MI455X (gfx1250) — compile-verified
//
#include <hip/hip_runtime.h>
#include <math.h>
#include <stdint.h>

#define B_ 8
#define H_ 128
#define W_ 128
#define C_ 20
#define G_ 64
#define TPB 256
#define TILES ((H_*W_)/TPB)            /* 64  */
#define NBLK  (B_*TILES)               /* 512 */

/* workspace layout (dwords) */
#define MAIN_OFF 0                     /* B*G*8  = 4096 */
#define SORT_OFF (MAIN_OFF + B_*G_*8)  /* 4096; B*G*4 = 2048 */
#define CNT_OFF  (SORT_OFF + B_*G_*4)  /* 6144; B*32  = 256  */
#define NV_OFF   (CNT_OFF  + B_*32)    /* 6400; 16 */
#define PART_OFF (NV_OFF   + 16)       /* 6416; + 512*6 = 9488 dwords total */

#define USE_ASYNC_STAGE 1

__device__ __forceinline__ float sigmoidf_(float x) { return 1.0f / (1.0f + expf(-x)); }
__device__ __forceinline__ float logsigf_(float x) {
  return (x >= 0.0f) ? -log1pf(expf(-x)) : (x - log1pf(expf(x)));
}

/* -------- kernel 0: per-GT prep, argmin slice, class-counting-sort -------- */
__global__ void cent_prep(const float* __restrict__ gt,
                          const float* __restrict__ sy,
                          const float* __restrict__ sx,
                          float* __restrict__ wsf)
{
  const int b = blockIdx.x, g = threadIdx.x;       /* 64 threads */
  __shared__ float cys[G_];
  __shared__ float squad[G_ * 4];                  /* fgy,fgx,isy,isx per gt */
  __shared__ int   scid[G_];
  __shared__ int   s_slice;
  __shared__ int   lcnt[C_ + 1];
  __shared__ int   lcur[C_];

  const float* row = gt + (b * G_ + g) * 7;
  float cy = row[0];
  cys[g] = cy;
  __syncthreads();
  if (g == 0) {                                    /* first-occurrence argmin */
    float mn = cys[0]; int idx = 0;
    for (int i = 1; i < G_; ++i) if (cys[i] < mn) { mn = cys[i]; idx = i; }
    s_slice = idx;
    wsf[NV_OFF + b] = fmaxf((float)idx, 1.0f);     /* n_valid */
  }
  __syncthreads();
  const int slice = s_slice;
  float vf  = (g < slice) ? 1.0f : 0.0f;
  float cx  = row[1];
  float by1 = (row[2] + 0.5f) * 0.25f;
  float bx1 = (row[3] + 0.5f) * 0.25f;
  float by2 = (row[4] + 0.5f) * 0.25f;
  float bx2 = (row[5] + 0.5f) * 0.25f;
  float gy  = (cy + 0.5f) * 0.25f;
  float gxx = (cx + 0.5f) * 0.25f;
  int cid = (int)row[6] - 1;
  int kpy = (int)floorf(cy * 0.25f); kpy = kpy < 0 ? 0 : (kpy > H_-1 ? H_-1 : kpy);
  int kpx = (int)floorf(cx * 0.25f); kpx = kpx < 0 ? 0 : (kpx > W_-1 ? W_-1 : kpx);
  int kpl = (vf > 0.0f && cid >= 0 && cid < C_) ? (kpy * W_ + kpx) : -1;
  float syv = sy[b * G_ + g], sxv = sx[b * G_ + g];

  float* o = wsf + MAIN_OFF + (b * G_ + g) * 8;
  o[0] = bx1; o[1] = bx2; o[2] = by1; o[3] = by2;
  o[4] = vf;  o[5] = __int_as_float(cid);
  o[6] = __int_as_float(kpl); o[7] = 0.0f;

  squad[g*4+0] = floorf(gy);  squad[g*4+1] = floorf(gxx);
  squad[g*4+2] = 1.0f / syv;  squad[g*4+3] = 1.0f / sxv;
  scid[g] = cid;
  __syncthreads();

  if (g == 0) {                                    /* serial counting sort by class */
    for (int c = 0; c <= C_; ++c) lcnt[c] = 0;
    for (int i = 0; i < slice; ++i) {
      int c = scid[i];
      if (c >= 0 && c < C_) lcnt[c]++;
    }
    int acc = 0;
    int* cw = (int*)wsf + CNT_OFF + b * 32;
    for (int c = 0; c < C_; ++c) { int n = lcnt[c]; cw[c] = acc; lcur[c] = acc; acc += n; }
    cw[C_] = acc;
    for (int c = C_ + 1; c < 32; ++c) cw[c] = 0;
    float* sb = wsf + SORT_OFF + b * (G_ * 4);
    for (int i = 0; i < slice; ++i) {
      int c = scid[i];
      if (c >= 0 && c < C_) {
        int pos = lcur[c]++;
        sb[pos*4+0] = squad[i*4+0]; sb[pos*4+1] = squad[i*4+1];
        sb[pos*4+2] = squad[i*4+2]; sb[pos*4+3] = squad[i*4+3];
      }
    }
  }
}

/* ---------------- kernel 1: one pixel per thread ---------------- */
__global__ void __launch_bounds__(TPB) cent_pixels(
    const float* __restrict__ kpts, const float* __restrict__ preg,
    const float* __restrict__ fpn,  const float* __restrict__ masks,
    const float* __restrict__ wsf,  float* __restrict__ partials)
{
  __shared__ __align__(16) float smain[G_ * 8];    /* 2048 B */
  __shared__ __align__(16) float ssort[G_ * 4];    /* 1024 B */
  __shared__ __align__(16) int   scnt[32];         /* 128 B  */
  __shared__ float rbuf[TPB];
  const int b    = blockIdx.x / TILES;
  const int tile = blockIdx.x % TILES;
  const int tid  = threadIdx.x;
  const float* gmain = wsf + MAIN_OFF + b * (G_ * 8);
  const float* gsort = wsf + SORT_OFF + b * (G_ * 4);
  const float* gcnt  = wsf + CNT_OFF  + b * 32;

#if USE_ASYNC_STAGE
  /* CDNA5 async global->LDS staging: 128 + 64 + 8 b128 transfers */
  if (tid < 128) {
    const float* gp = gmain + tid * 4;
    unsigned la = (unsigned)(uintptr_t)(&smain[tid * 4]);
    asm volatile("global_load_async_to_lds_b128 %0, %1, off" :: "v"(la), "v"(gp) : "memory");
  } else if (tid < 192) {
    const float* gp = gsort + (tid - 128) * 4;
    unsigned la = (unsigned)(uintptr_t)(&ssort[(tid - 128) * 4]);
    asm volatile("global_load_async_to_lds_b128 %0, %1, off" :: "v"(la), "v"(gp) : "memory");
  } else if (tid < 200) {
    const float* gp = gcnt + (tid - 192) * 4;
    unsigned la = (unsigned)(uintptr_t)(&scnt[(tid - 192) * 4]);
    asm volatile("global_load_async_to_lds_b128 %0, %1, off" :: "v"(la), "v"(gp) : "memory");
  }
  asm volatile("s_wait_asynccnt 0x0" ::: "memory");
  __syncthreads();
#else
  for (int i = tid; i < G_ * 8; i += TPB) smain[i] = gmain[i];
  for (int i = tid; i < G_ * 4; i += TPB) ssort[i] = gsort[i];
  if (tid < 32) scnt[tid] = ((const int*)gcnt)[tid];
  __syncthreads();
#endif

  const int p = tile * TPB + tid;
  const int h = p >> 7, w = p & (W_ - 1);
  const float fh = (float)h, fw = (float)w;
  const float gyc = fh + 0.5f, gxc = fw + 0.5f;
  const size_t pix = (size_t)((b * H_ + h) * W_ + w);
  const float* mrow = masks + pix * G_;
  const float* krow = kpts  + pix * C_;
  const float* frow = fpn   + pix * C_;
  const float* prow = preg  + pix * 4;
  __builtin_prefetch(krow, 0, 1);                  /* global_prefetch_b8 */
  __builtin_prefetch(frow, 0, 1);
  __builtin_prefetch(prow, 0, 1);

  float loc  = -3.402823466e38f;
  float amin =  3.402823466e38f;
  unsigned kpbits = 0;

  /* ---- pass 1: area_min, loc, gt_kp bits (float4-vectorized mask reads) ---- */
  for (int g4 = 0; g4 < G_; g4 += 4) {
    float4 m4 = *(const float4*)(mrow + g4);
    #pragma unroll
    for (int k = 0; k < 4; ++k) {
      const int g = g4 + k;
      const float m = (k == 0) ? m4.x : (k == 1) ? m4.y : (k == 2) ? m4.z : m4.w;
      const float* r = &smain[g * 8];
      float bx1 = r[0], bx2 = r[1], by1 = r[2], by2 = r[3], vf = r[4];
      int cid = __float_as_int(r[5]);
      int kpl = __float_as_int(r[6]);
      float dl0 = gxc - bx1, dr0 = bx2 - gxc;
      float dt0 = gyc - by1, db0 = by2 - gyc;
      bool ind = (dt0 > 0.0f) && (db0 > 0.0f) && (dl0 > 0.0f) && (dr0 > 0.0f);
      float heat = ind ? __fmul_rn(m, vf) : 0.0f;
      float dl = __fmul_rn(dl0, heat), dr = __fmul_rn(dr0, heat);
      float dt = __fmul_rn(dt0, heat), db = __fmul_rn(db0, heat);
      float area = __fmul_rn(__fadd_rn(dl, dr), __fadd_rn(dt, db));
      float apad = __fadd_rn(area, __fmul_rn(__fsub_rn(1.0f, heat), 1e8f));
      amin = fminf(amin, apad);
      loc  = fmaxf(loc, heat);
      if (kpl == p) kpbits |= (1u << (cid & 31));
    }
  }

  /* ---- pass R: class-sorted min-z -> red_gt[c] = exp(-min z), iou_red ---- */
  float redgt[C_];
  float minzg = 3.402823466e38f;
  #pragma unroll
  for (int c = 0; c < C_; ++c) {
    const int s0 = scnt[c], e0 = scnt[c + 1];
    float mz = 3.402823466e38f;
    for (int i = s0; i < e0; ++i) {
      const float* q = &ssort[i * 4];
      float dy = fh - q[0], dx = fw - q[1];
      float z = 0.5f * (dy * dy * q[2] + dx * dx * q[3]);
      mz = fminf(mz, z);
    }
    redgt[c] = (e0 > s0) ? expf(-mz) : 0.0f;       /* empty class -> 0 */
    minzg = fminf(minzg, mz);
  }
  float ired = (scnt[C_] > 0) ? expf(-minzg) : 0.0f;

  /* ---- pass 2: exact-equality assignment, dl/dr/dt/db, hm_gt bits ---- */
  float dlm = -3.402823466e38f, drm = -3.402823466e38f;
  float dtm = -3.402823466e38f, dbm = -3.402823466e38f;
  unsigned hmbits = 0;
  for (int g4 = 0; g4 < G_; g4 += 4) {
    float4 m4 = *(const float4*)(mrow + g4);
    #pragma unroll
    for (int k = 0; k < 4; ++k) {
      const int g = g4 + k;
      const float m = (k == 0) ? m4.x : (k == 1) ? m4.y : (k == 2) ? m4.z : m4.w;
      const float* r = &smain[g * 8];
      float bx1 = r[0], bx2 = r[1], by1 = r[2], by2 = r[3], vf = r[4];
      int cid = __float_as_int(r[5]);
      float dl0 = gxc - bx1, dr0 = bx2 - gxc;
      float dt0 = gyc - by1, db0 = by2 - gyc;
      bool ind = (dt0 > 0.0f) && (db0 > 0.0f) && (dl0 > 0.0f) && (dr0 > 0.0f);
      float heat = ind ? __fmul_rn(m, vf) : 0.0f;
      float dl = __fmul_rn(dl0, heat), dr = __fmul_rn(dr0, heat);
      float dt = __fmul_rn(dt0, heat), db = __fmul_rn(db0, heat);
      float area = __fmul_rn(__fadd_rn(dl, dr), __fadd_rn(dt, db));
      bool eq = (area == amin);
      float dm = eq ? loc : 0.0f;                  /* dist_mask value = eq*loc */
      dlm = fmaxf(dlm, __fmul_rn(dl, dm));
      drm = fmaxf(drm, __fmul_rn(dr, dm));
      dtm = fmaxf(dtm, __fmul_rn(dt, dm));
      dbm = fmaxf(dbm, __fmul_rn(db, dm));
      if (eq && vf > 0.0f && cid >= 0 && cid < C_) hmbits |= (1u << cid);
    }
  }

  /* ---- IoU loss ---- */
  float pl = prow[0], pr = prow[1], pt = prow[2], pb = prow[3];
  float inter = (fminf(dlm, pl) + fminf(drm, pr)) * (fminf(dtm, pt) + fminf(dbm, pb));
  float uni   = (dlm + drm) * (dtm + dbm) + (pl + pr) * (pt + pb) - inter;
  float iou   = inter / (uni + 1e-12f);
  float iou_loss = -logf(iou + 1e-12f) * loc * (ired * 4.0f + 1.0f);

  /* ---- per-class focal terms ---- */
  float grav = 0.0f, hpos = 0.0f, hneg = 0.0f, shg = 0.0f;
  float locpos = fmaxf(loc, 0.0f);
  #pragma unroll
  for (int c = 0; c < C_; ++c) {
    float gp = krow[c], hp = frow[c];
    float sg = sigmoidf_(gp), lsg = logsigf_(gp);
    float kpv = (float)((kpbits >> c) & 1u);
    float rg = redgt[c];
    float oms = 1.0f - sg;
    float t = 1.0f - rg; float t4 = (t * t) * (t * t);
    grav += -(oms * oms) * lsg * kpv
          + -t4 * (sg * sg) * (lsg - gp) * (1.0f - kpv);
    float hm = ((hmbits >> c) & 1u) ? locpos : 0.0f;
    float sh = sigmoidf_(hp), lsh = logsigf_(hp);
    float omh = 1.0f - sh;
    hpos += -10.0f * (omh * omh) * lsh * hm;
    hneg += -10.0f * (sh * sh) * (lsh - hp) * (1.0f - hm);
    shg  += hm;
  }

  /* ---- deterministic block tree-reduce of 6 partial sums ---- */
  float vals[6] = { iou_loss, grav, hneg, hpos, shg, loc };
  #pragma unroll
  for (int q = 0; q < 6; ++q) {
    __syncthreads();
    rbuf[tid] = vals[q];
    __syncthreads();
    for (int s = TPB / 2; s > 0; s >>= 1) {
      if (tid < s) rbuf[tid] += rbuf[tid + s];
      __syncthreads();
    }
    if (tid == 0) partials[blockIdx.x * 6 + q] = rbuf[0];
  }
}

/* ---------------- kernel 2: fixed-order final reduction ---------------- */
__global__ void cent_final(const float* __restrict__ wsf, float* __restrict__ out)
{
  __shared__ float tot[B_];
  const int t = threadIdx.x;
  if (t < B_) {
    float s0 = 0, s1 = 0, s2 = 0, s3 = 0, s4 = 0, s5 = 0;
    const float* pp = wsf + PART_OFF + (t * TILES) * 6;
    for (int k = 0; k < TILES; ++k) {
      s0 += pp[k * 6 + 0]; s1 += pp[k * 6 + 1]; s2 += pp[k * 6 + 2];
      s3 += pp[k * 6 + 3]; s4 += pp[k * 6 + 4]; s5 += pp[k * 6 + 5];
    }
    float nv   = wsf[NV_OFF + t];
    float shg  = s4;
    float hm_loss = s2 / ((float)(H_ * W_ * C_) - shg);
    bool  cond = (shg != 0.0f);
    float safe_hg  = cond ? shg : 1.0f;
    float safe_loc = (s5 > 0.0f) ? s5 : 1.0f;
    float hm2      = hm_loss + s3 / safe_hg;
    float iou_out  = cond ? (s0 / safe_loc) : 0.0f;
    float grav_out = cond ? (s1 / nv) : 0.0f;
    tot[t] = cond ? (iou_out + grav_out + hm2) : hm_loss;
  }
  __syncthreads();
  if (t == 0) {
    float s = 0.0f;
    for (int i = 0; i < B_; ++i) s += tot[i];
    out[0] = s * 0.125f;
  }
}

extern "C" void kernel_launch(void* const* d_in, const int* in_sizes, int n_in,
                              void* d_out, int out_size, void* d_ws, size_t ws_size,
                              hipStream_t stream)
{
  const float* kpts = (const float*)d_in[0];
  const float* preg = (const float*)d_in[1];
  const float* fpn  = (const float*)d_in[2];
  const float* gt   = (const float*)d_in[3];
  const float* msk  = (const float*)d_in[4];
  const float* sy   = (const float*)d_in[5];
  const float* sx   = (const float*)d_in[6];
  float* wsf = (float*)d_ws;
  float* partials = wsf + PART_OFF;

  cent_prep  <<<B_,   G_,  0, stream>>>(gt, sy, sx, wsf);
  cent_pixels<<<NBLK, TPB, 0, stream>>>(kpts, preg, fpn, msk, wsf, partials);
  cent_final <<<1,    64,  0, stream>>>(wsf, (float*)d_out);
}